// TemporalTransformerConv_36584531427428
// MI455X (gfx1250) — compile-verified
//
#include <hip/hip_runtime.h>
#include <math.h>

// ---------------------------------------------------------------------------
// TemporalTransformerConv (TGAT-style edge-GAT) — algebraically reduced.
//
// Key reduction: node_feat/edge_feat are only consumed via per-head dot
// products with attn_{l,r,e}, so fc_node_w/fc_edge_w collapse to rank-reduced
// matrices Wn[100x4] (el_h0, el_h1, er_h0, er_h1) and We[200x2] (ee_h0, ee_h1).
// Pipeline (all HBM-bound, ~400MB total => ~17us at 23.3 TB/s):
//   K0  reduce weights (tiny)
//   K1  el/er = memory @ Wn          -- WMMA f32 16x16x4, 16 nodes/wave
//   K2  ee    = [raw|cos(te)] @ We   -- WMMA f32 16x16x4, 16 edges/wave,
//       then logits + atomicMax per (dst,head) (monotone-uint encoding)
//   K3  finalize segment max (zero-in-degree -> 0)
//   K4  exp + atomicAdd numerator/denominator per (dst,head)
//   K5  ft = num/den ; out = memory + 0.5*(ft0+ft1)
// ---------------------------------------------------------------------------

#define N_NODES 50000
#define N_EDGES 800000
#define NODE_F  100
#define EDGE_F  100
#define TIME_D  100
#define HEADS   2
#define OUT_F   100
#define NEG_SLOPE 0.2f

typedef __attribute__((ext_vector_type(2))) float v2f;
typedef __attribute__((ext_vector_type(8))) float v8f;

#if __has_builtin(__builtin_amdgcn_wmma_f32_16x16x4_f32)
#define HAVE_WMMA_F32X4 1
#else
#define HAVE_WMMA_F32X4 0
#endif

// ---- workspace layout (float units) ----
// wnT : [16][104] node reduced weights, col-major (col0/1=el h0/h1, col2/3=er)
// weT : [16][200] edge reduced weights, col-major (col0/1=ee h0/h1)
#define OFF_WNT   0u
#define OFF_NBIAS 1664u
#define OFF_WET   1680u
#define OFF_EBIAS 4880u
#define OFF_ELR   4896u            // [N][4] : el0 el1 er0 er1
#define OFF_UMAX  204896u          // [N][2] uint, monotone-encoded max
#define OFF_EMAX  304896u          // [N][2] float
#define OFF_DEN   404896u          // [N][2]
#define OFF_NUM   504896u          // [N][2]
#define OFF_FTS   604896u          // [N]    0.5*(ft0+ft1)
#define OFF_EP    654896u          // [E][2] el_prime
#define OFF_LG    2254896u         // [E][2] leaky-relu logits
#define WS_FLOATS 3854896u

__device__ __forceinline__ unsigned fenc(float f) {
  unsigned u = __float_as_uint(f);
  return (u & 0x80000000u) ? ~u : (u | 0x80000000u);   // monotone map
}
__device__ __forceinline__ float fdec(unsigned u) {
  unsigned b = (u & 0x80000000u) ? (u & 0x7fffffffu) : ~u;
  return __uint_as_float(b);
}

// ---------------- K0: reduce weights ----------------
__global__ void k_weights(const float* __restrict__ fc_node_w,
                          const float* __restrict__ fc_node_b,
                          const float* __restrict__ fc_edge_w,
                          const float* __restrict__ fc_edge_b,
                          const float* __restrict__ attn_l,
                          const float* __restrict__ attn_r,
                          const float* __restrict__ attn_e,
                          float* __restrict__ ws) {
  float* wnT   = ws + OFF_WNT;
  float* nbias = ws + OFF_NBIAS;
  float* weT   = ws + OFF_WET;
  float* ebias = ws + OFF_EBIAS;
  int t = threadIdx.x;
  for (int i = t; i < 16 * 104; i += blockDim.x) wnT[i] = 0.f;
  for (int i = t; i < 16 * 200; i += blockDim.x) weT[i] = 0.f;
  if (t < 16) { nbias[t] = 0.f; ebias[t] = 0.f; }
  __syncthreads();
  // node: 4 cols x 100 k
  for (int job = t; job < 400; job += blockDim.x) {
    int col = job / 100, k = job % 100, h = col & 1;
    const float* attn = (col < 2) ? attn_l : attn_r;
    float s = 0.f;
    for (int f = 0; f < OUT_F; ++f)
      s += fc_node_w[k * (HEADS * OUT_F) + h * OUT_F + f] * attn[h * OUT_F + f];
    wnT[col * 104 + k] = s;
  }
  // edge: 2 cols x 200 k
  for (int job = t; job < 400; job += blockDim.x) {
    int h = job / 200, k = job % 200;
    float s = 0.f;
    for (int f = 0; f < OUT_F; ++f)
      s += fc_edge_w[k * (HEADS * OUT_F) + h * OUT_F + f] * attn_e[h * OUT_F + f];
    weT[h * 200 + k] = s;
  }
  if (t < 4) {
    int h = t & 1;
    const float* attn = (t < 2) ? attn_l : attn_r;
    float s = 0.f;
    for (int f = 0; f < OUT_F; ++f) s += fc_node_b[h * OUT_F + f] * attn[h * OUT_F + f];
    nbias[t] = s;
  } else if (t < 6) {
    int h = t - 4;
    float s = 0.f;
    for (int f = 0; f < OUT_F; ++f) s += fc_edge_b[h * OUT_F + f] * attn_e[h * OUT_F + f];
    ebias[h] = s;
  }
}

// ---------------- K_init: zero accumulators ----------------
__global__ void k_init(float* __restrict__ ws) {
  unsigned i = blockIdx.x * blockDim.x + threadIdx.x;
  if (i < 2u * N_NODES) {
    ((unsigned*)(ws + OFF_UMAX))[i] = 0u;   // == fenc of "untouched" sentinel
    (ws + OFF_DEN)[i] = 0.f;
    (ws + OFF_NUM)[i] = 0.f;
  }
}

#if HAVE_WMMA_F32X4
// ---------------- K1: el/er via WMMA (16 nodes per wave) ----------------
__global__ __launch_bounds__(128) void k_node(const float* __restrict__ memory,
                                              float* __restrict__ ws) {
  const float* wnT   = ws + OFF_WNT;
  const float* nbias = ws + OFF_NBIAS;
  float*       elr   = ws + OFF_ELR;
  int wave = blockIdx.x * (blockDim.x >> 5) + (threadIdx.x >> 5);
  if (wave >= N_NODES / 16) return;                    // 3125 tiles, exact
  int lane = threadIdx.x & 31, half = lane >> 4, lcol = lane & 15;
  int r0 = wave * 16;
  const float* arow = memory + (unsigned)(r0 + lcol) * NODE_F;  // lane's A row
  const float* bcol = wnT + lcol * 104;                         // lane's B col
  v8f acc = {0.f, 0.f, 0.f, 0.f, 0.f, 0.f, 0.f, 0.f};
#pragma unroll
  for (int kk = 0; kk < NODE_F; kk += 4) {             // K = 100 = 25 x 4
    int k0 = kk + 2 * half;
    v2f a = *(const v2f*)(arow + k0);
    v2f b = *(const v2f*)(bcol + k0);
    acc = __builtin_amdgcn_wmma_f32_16x16x4_f32(false, a, false, b,
                                                (short)0, acc, false, false);
  }
  if (lcol < 4) {                                      // cols 0..3 are live
    float bias = nbias[lcol];
#pragma unroll
    for (int j = 0; j < 8; ++j) {
      int row = j + 8 * half;
      elr[(unsigned)(r0 + row) * 4 + lcol] = acc[j] + bias;
    }
  }
}

// ---------------- K2: ee via WMMA + logits + segment max ----------------
__global__ __launch_bounds__(128) void k_edge(const float* __restrict__ efr,
                                              const float* __restrict__ edge_ts,
                                              const float* __restrict__ node_ts,
                                              const int* __restrict__ src,
                                              const int* __restrict__ dst,
                                              const float* __restrict__ time_w,
                                              const float* __restrict__ time_b,
                                              float* __restrict__ ws) {
  const float* weT   = ws + OFF_WET;
  const float* ebias = ws + OFF_EBIAS;
  const float* elr   = ws + OFF_ELR;
  unsigned*    umax  = (unsigned*)(ws + OFF_UMAX);
  float*       ep    = ws + OFF_EP;
  float*       lg    = ws + OFF_LG;
  int wave = blockIdx.x * (blockDim.x >> 5) + (threadIdx.x >> 5);
  if (wave >= N_EDGES / 16) return;                    // 50000 tiles, exact
  int lane = threadIdx.x & 31, half = lane >> 4, lcol = lane & 15;
  unsigned e0 = (unsigned)wave * 16;
  unsigned my_e = e0 + lcol;
  float td = edge_ts[my_e] - node_ts[src[my_e]];       // lane's row time-delta
  const float* arow = efr + (size_t)my_e * EDGE_F;
  const float* bcol = weT + lcol * 200;
  v8f acc = {0.f, 0.f, 0.f, 0.f, 0.f, 0.f, 0.f, 0.f};
#pragma unroll
  for (int kk = 0; kk < EDGE_F + TIME_D; kk += 4) {    // K = 200 = 50 x 4
    int k0 = kk + 2 * half;
    v2f a;
    if (k0 < EDGE_F) {                                 // raw edge features
      a = *(const v2f*)(arow + k0);
    } else {                                           // Fourier time encoding
      int j = k0 - EDGE_F;
      a.x = __cosf(td * time_w[j] + time_b[j]);
      a.y = __cosf(td * time_w[j + 1] + time_b[j + 1]);
    }
    v2f b = *(const v2f*)(bcol + k0);
    acc = __builtin_amdgcn_wmma_f32_16x16x4_f32(false, a, false, b,
                                                (short)0, acc, false, false);
  }
  if (lcol < HEADS) {                                  // cols 0..1 = heads
    int h = lcol;
    float eb = ebias[h];
#pragma unroll
    for (int j = 0; j < 8; ++j) {
      unsigned e = e0 + j + 8 * half;
      float ee = acc[j] + eb;
      int s = src[e], d = dst[e];
      float elp   = elr[(unsigned)s * 4 + h] + ee;            // u_add_e
      float logit = elp + elr[(unsigned)d * 4 + 2 + h];       // e_add_v
      logit = logit > 0.f ? logit : NEG_SLOPE * logit;        // leaky relu
      ep[(size_t)e * 2 + h] = elp;
      lg[(size_t)e * 2 + h] = logit;
      atomicMax(&umax[(unsigned)d * 2 + h], fenc(logit));
    }
  }
}
#else
// ---------------- scalar fallbacks (only if WMMA builtin absent) ----------
__global__ void k_node(const float* __restrict__ memory, float* __restrict__ ws) {
  int n = blockIdx.x * blockDim.x + threadIdx.x;
  if (n >= N_NODES) return;
  const float* wnT = ws + OFF_WNT;
  const float* nbias = ws + OFF_NBIAS;
  float acc[4] = {0, 0, 0, 0};
  for (int k = 0; k < NODE_F; ++k) {
    float m = memory[(unsigned)n * NODE_F + k];
    for (int c = 0; c < 4; ++c) acc[c] += m * wnT[c * 104 + k];
  }
  for (int c = 0; c < 4; ++c) (ws + OFF_ELR)[(unsigned)n * 4 + c] = acc[c] + nbias[c];
}
__global__ void k_edge(const float* __restrict__ efr, const float* __restrict__ edge_ts,
                       const float* __restrict__ node_ts, const int* __restrict__ src,
                       const int* __restrict__ dst, const float* __restrict__ time_w,
                       const float* __restrict__ time_b, float* __restrict__ ws) {
  unsigned e = blockIdx.x * blockDim.x + threadIdx.x;
  if (e >= N_EDGES) return;
  const float* weT = ws + OFF_WET;
  const float* elr = ws + OFF_ELR;
  int s = src[e], d = dst[e];
  float td = edge_ts[e] - node_ts[s];
  float ee0 = (ws + OFF_EBIAS)[0], ee1 = (ws + OFF_EBIAS)[1];
  for (int k = 0; k < EDGE_F + TIME_D; ++k) {
    float a = (k < EDGE_F) ? efr[(size_t)e * EDGE_F + k]
                           : __cosf(td * time_w[k - EDGE_F] + time_b[k - EDGE_F]);
    ee0 += a * weT[k];
    ee1 += a * weT[200 + k];
  }
  for (int h = 0; h < HEADS; ++h) {
    float elp = elr[(unsigned)s * 4 + h] + (h ? ee1 : ee0);
    float logit = elp + elr[(unsigned)d * 4 + 2 + h];
    logit = logit > 0.f ? logit : NEG_SLOPE * logit;
    (ws + OFF_EP)[(size_t)e * 2 + h] = elp;
    (ws + OFF_LG)[(size_t)e * 2 + h] = logit;
    atomicMax((unsigned*)(ws + OFF_UMAX) + (unsigned)d * 2 + h, fenc(logit));
  }
}
#endif

// ---------------- K3: finalize segment max ----------------
__global__ void k_emax(float* __restrict__ ws) {
  unsigned i = blockIdx.x * blockDim.x + threadIdx.x;
  if (i < 2u * N_NODES) {
    unsigned u = ((const unsigned*)(ws + OFF_UMAX))[i];
    (ws + OFF_EMAX)[i] = (u == 0u) ? 0.f : fdec(u);    // zero-in-degree -> 0
  }
}

// ---------------- K4: exp + segment sums (denominator & numerator) -------
__global__ void k_softnum(const int* __restrict__ dst, float* __restrict__ ws) {
  unsigned i = blockIdx.x * blockDim.x + threadIdx.x;
  if (i >= 2u * N_EDGES) return;
  unsigned h = i & 1u, e = i >> 1;
  int d = dst[e];
  float x = __expf((ws + OFF_LG)[i] - (ws + OFF_EMAX)[(unsigned)d * 2 + h]);
  atomicAdd(ws + OFF_DEN + (unsigned)d * 2 + h, x);
  atomicAdd(ws + OFF_NUM + (unsigned)d * 2 + h, x * (ws + OFF_EP)[i]);
}

// ---------------- K4b: ft = num/den, averaged over heads ----------------
__global__ void k_ft(float* __restrict__ ws) {
  unsigned n = blockIdx.x * blockDim.x + threadIdx.x;
  if (n >= N_NODES) return;
  float d0 = (ws + OFF_DEN)[n * 2], d1 = (ws + OFF_DEN)[n * 2 + 1];
  float f0 = d0 > 0.f ? (ws + OFF_NUM)[n * 2] / d0 : 0.f;
  float f1 = d1 > 0.f ? (ws + OFF_NUM)[n * 2 + 1] / d1 : 0.f;
  (ws + OFF_FTS)[n] = 0.5f * (f0 + f1);
}

// ---------------- K5: out = memory + mean_h(ft) ----------------
__global__ void k_out(const float* __restrict__ memory, const float* __restrict__ ws,
                      float* __restrict__ out) {
  unsigned i = blockIdx.x * blockDim.x + threadIdx.x;
  if (i < (unsigned)N_NODES * NODE_F)
    out[i] = memory[i] + (ws + OFF_FTS)[i / NODE_F];
}

extern "C" void kernel_launch(void* const* d_in, const int* in_sizes, int n_in,
                              void* d_out, int out_size, void* d_ws, size_t ws_size,
                              hipStream_t stream) {
  const float* memory    = (const float*)d_in[0];
  const float* node_ts   = (const float*)d_in[1];
  const float* efr       = (const float*)d_in[2];
  const float* edge_ts   = (const float*)d_in[3];
  const float* time_w    = (const float*)d_in[4];
  const float* time_b    = (const float*)d_in[5];
  const float* fc_node_w = (const float*)d_in[6];
  const float* fc_node_b = (const float*)d_in[7];
  const float* fc_edge_w = (const float*)d_in[8];
  const float* fc_edge_b = (const float*)d_in[9];
  const float* attn_l    = (const float*)d_in[10];
  const float* attn_r    = (const float*)d_in[11];
  const float* attn_e    = (const float*)d_in[12];
  const int*   src       = (const int*)d_in[13];
  const int*   dst       = (const int*)d_in[14];
  float* out = (float*)d_out;
  float* ws  = (float*)d_ws;
  if (ws_size < (size_t)WS_FLOATS * sizeof(float)) return;

  k_weights<<<1, 256, 0, stream>>>(fc_node_w, fc_node_b, fc_edge_w, fc_edge_b,
                                   attn_l, attn_r, attn_e, ws);
  k_init<<<(2 * N_NODES + 255) / 256, 256, 0, stream>>>(ws);

#if HAVE_WMMA_F32X4
  // one wave per 16-row tile, 4 waves per block
  k_node<<<(N_NODES / 16 + 3) / 4, 128, 0, stream>>>(memory, ws);
  k_edge<<<(N_EDGES / 16) / 4, 128, 0, stream>>>(efr, edge_ts, node_ts, src, dst,
                                                 time_w, time_b, ws);
#else
  k_node<<<(N_NODES + 255) / 256, 256, 0, stream>>>(memory, ws);
  k_edge<<<(N_EDGES + 255) / 256, 256, 0, stream>>>(efr, edge_ts, node_ts, src, dst,
                                                    time_w, time_b, ws);
#endif

  k_emax<<<(2 * N_NODES + 255) / 256, 256, 0, stream>>>(ws);
  k_softnum<<<(2 * N_EDGES + 255) / 256, 256, 0, stream>>>(dst, ws);
  k_ft<<<(N_NODES + 255) / 256, 256, 0, stream>>>(ws);
  k_out<<<(N_NODES * NODE_F + 255) / 256, 256, 0, stream>>>(memory, ws, out);
}